// SequentialPendul_start_8710193677064
// MI455X (gfx1250) — compile-verified
//
#include <hip/hip_runtime.h>
#include <stdint.h>

// Problem constants (match reference)
#define NBCH 131072   // number of chains
#define NLNK 64       // links per chain
#define FDIM 4        // feature columns (float4 per (chain, link))

// Tiling
#define CH   32               // chains staged per block (32 KiB payload)
#define BLK  256              // threads per block (8 waves)
#define ROWQ (NLNK + 1)       // padded LDS row stride in float4 units (65)

typedef __attribute__((ext_vector_type(4))) unsigned int u32x4;
typedef __attribute__((ext_vector_type(4))) int          i32x4;
typedef __attribute__((ext_vector_type(8))) int          i32x8;

__global__ __launch_bounds__(BLK)
void pendulum_scan_kernel(const float* __restrict__ x,
                          const float* __restrict__ r_vec,
                          float* __restrict__ out) {
  // Padded tile: CH rows of (NLNK + 1 pad) float4  -> 33,280 bytes
  __shared__ float4 tile[CH * ROWQ];
  __shared__ float  rbuf[NLNK];

  const int      tid   = threadIdx.x;
  const unsigned blk   = blockIdx.x;
  const size_t   base4 = (size_t)blk * (CH * NLNK);   // block's first float4
  const float4* __restrict__ x4 = (const float4*)x;
  float4* __restrict__       o4 = (float4*)out;

#if defined(__has_builtin) && __has_builtin(__builtin_amdgcn_tensor_load_to_lds) && __has_builtin(__builtin_amdgcn_s_wait_tensorcnt)
  // ---- TDM load: one tensor_load_to_lds DMAs the whole 32 KiB chunk, and
  // D# padding (+4 DWORDs after every 256 DWORDs) produces the 65-float4
  // row stride in LDS (one pad float4 after each chain's 64 float4s).
  if (tid < 32) {  // wave 0 only: exactly one TDM op per block
    const unsigned long long gaddr =
        (unsigned long long)(const void*)(x4 + base4);
    const unsigned lds_base =
        (unsigned)(unsigned long long)(const void*)&tile[0];

    const unsigned ndw = CH * NLNK * FDIM;  // 8192 DWORDs in the tile

    u32x4 g0;
    g0.x = 1u;                                        // count=1, user mode
    g0.y = lds_base;                                  // lds_addr (bytes)
    g0.z = (unsigned)(gaddr & 0xFFFFFFFFull);         // global_addr[31:0]
    g0.w = (unsigned)((gaddr >> 32) & 0x01FFFFFFull)  // global_addr[56:32]
         | (2u << 30);                                // type = 2

    i32x8 g1;
    g1[0] = (int)((2u << 16)    // data_size = 4 bytes
                | (1u << 20)    // pad_enable
                | (7u << 22)    // pad_interval: 256 DWORDs
                | (3u << 25));  // pad_amount: 4 DWORDs (one float4)
    g1[1] = (int)((ndw & 0xFFFFu) << 16);             // tensor_dim0[15:0]
    g1[2] = (int)(((ndw >> 16) & 0xFFFFu)             // tensor_dim0[31:16]
                | (1u << 16));                        // tensor_dim1 = 1
    g1[3] = (int)((ndw & 0xFFFFu) << 16);             // tile_dim0 = ndw
    g1[4] = 0;                                        // tile_dim1/2 unused
    g1[5] = (int)ndw;                                 // dim0_stride[31:0]
    g1[6] = 0;
    g1[7] = 0;

    i32x4 z4 = {0, 0, 0, 0};
#if __clang_major__ >= 23
    i32x8 z8 = {0, 0, 0, 0, 0, 0, 0, 0};
    __builtin_amdgcn_tensor_load_to_lds(g0, g1, z4, z4, z8, 0);
#else
    __builtin_amdgcn_tensor_load_to_lds(g0, g1, z4, z4, 0);
#endif
  }
  if (tid < NLNK) rbuf[tid] = r_vec[tid];             // stage link lengths
  if (tid < 32) __builtin_amdgcn_s_wait_tensorcnt(0); // TENSORcnt == 0
  __syncthreads();                                    // publish LDS
#else
  // ---- Fallback: cooperative coalesced staging (also taken on host pass)
  for (int k = 0; k < (CH * NLNK) / BLK; ++k) {
    int g = tid + k * BLK;
    int c = g >> 6, i = g & 63;
    tile[c * ROWQ + i] = x4[base4 + g];
  }
  if (tid < NLNK) rbuf[tid] = r_vec[tid];
  __syncthreads();
#endif

  // ---- Serial per-chain scan: lane c of wave 0 owns chain c.
  // Output xy for step i equals the corrected anchor p, so write p back.
  if (tid < CH) {
    float4* row = &tile[tid * ROWQ];
    float px = 0.0f, py = 0.0f;
#pragma unroll 4
    for (int i = 0; i < NLNK; ++i) {
      float4 v  = row[i];
      float  dx = v.x - px;
      float  dy = v.y - py;
      float  s  = 1.0f - rbuf[i] / sqrtf(dx * dx + dy * dy);
      px = v.x - dx * s;   // = r_i - lam_i  (corrected position / next anchor)
      py = v.y - dy * s;
      row[i].x = px;       // x_new[:, :2]; z,w stay untouched in the tile
      row[i].y = py;
    }
  }
  __syncthreads();

  // ---- Coalesced write-out of the whole tile (all 8 waves).
  for (int k = 0; k < (CH * NLNK) / BLK; ++k) {
    int g = tid + k * BLK;
    int c = g >> 6, i = g & 63;
    o4[base4 + g] = tile[c * ROWQ + i];
  }
}

extern "C" void kernel_launch(void* const* d_in, const int* in_sizes, int n_in,
                              void* d_out, int out_size, void* d_ws, size_t ws_size,
                              hipStream_t stream) {
  const float* x     = (const float*)d_in[0];
  const float* r_vec = (const float*)d_in[1];
  // d_in[2] (batch, int64) is unused: chain index is implicit in the layout.
  float* out = (float*)d_out;
  (void)in_sizes; (void)n_in; (void)out_size; (void)d_ws; (void)ws_size;

  dim3 grid(NBCH / CH);   // 4096 blocks
  dim3 block(BLK);        // 256 threads = 8 waves
  hipLaunchKernelGGL(pendulum_scan_kernel, grid, block, 0, stream,
                     x, r_vec, out);
}